// QuanvolutionHybrid_65481071398272
// MI455X (gfx1250) — compile-verified
//
#include <hip/hip_runtime.h>
#include <math.h>

typedef __attribute__((ext_vector_type(2))) float v2f;
typedef __attribute__((ext_vector_type(8))) float v8f;

#define ROWS      16     // batch rows per block (WMMA M)
#define NPATCH    196
#define KFEAT     784
#define FPAD      788    // 788 % 64 == 20 -> conflict-free A-fragment reads
#define NTHREADS  128
#define NWAVES    4
#define KT_WAVE   49     // 196 k-tiles of 4 split across 4 waves
#define BPAD_FLOATS (KFEAT * 16)   // 12544 floats = 50,176 B

template <int WBIT>
__device__ __forceinline__ void apply_ry(float st[16], float c, float s) {
#pragma unroll
  for (int i = 0; i < 16; ++i) {
    if ((i & WBIT) == 0) {
      float a0 = st[i], a1 = st[i | WBIT];
      st[i]        = c * a0 - s * a1;
      st[i | WBIT] = s * a0 + c * a1;
    }
  }
}

template <int WC, int WT>
__device__ __forceinline__ void apply_cnot(float st[16]) {
#pragma unroll
  for (int i = 0; i < 16; ++i) {
    if ((i & WC) && !(i & WT)) {
      float t = st[i]; st[i] = st[i | WT]; st[i | WT] = t;
    }
  }
}

// Build zero-padded, transposed, K-pair-interleaved B:
//   Bpad[(k>>1)*32 + n*2 + (k&1)] = (n<10) ? lin_w[n*784 + k] : 0
// Consumer reads a v2f per lane: pair (k,k+1) for column n at 8-byte stride
// across lanes -> fully coalesced global_load_b64, no masking needed.
__global__ __launch_bounds__(256)
void prep_bpad_kernel(const float* __restrict__ lin_w, float* __restrict__ bpad) {
  int idx = blockIdx.x * 256 + threadIdx.x;   // over 784*16
  if (idx < BPAD_FLOATS) {
    int k = idx >> 4;
    int n = idx & 15;
    float v = (n < 10) ? lin_w[n * KFEAT + k] : 0.0f;
    bpad[((k >> 1) << 5) + (n << 1) + (k & 1)] = v;
  }
}

__global__ __launch_bounds__(NTHREADS)
void quanv_fused_kernel(const float* __restrict__ x,      // [B,784]
                        const float* __restrict__ wts,    // [3,4]
                        const float* __restrict__ lin_w,  // [10,784]
                        const float* __restrict__ lin_b,  // [10]
                        const float* __restrict__ bpad,   // [392][16][2] or unused
                        int useBpad,
                        float* __restrict__ out,          // [B,10]
                        int Btotal)
{
  __shared__ float sFeats[ROWS][FPAD];
  __shared__ float sC[NWAVES][16][16];

  const int tid = threadIdx.x;
  const int b0  = blockIdx.x * ROWS;

  // Uniform weight angles -> cos/sin of half-angles (scalar loads, cheap VALU)
  float cw[12], sw[12];
#pragma unroll
  for (int j = 0; j < 12; ++j) {
    float h = 0.5f * wts[j];
    cw[j] = __cosf(h);
    sw[j] = __sinf(h);
  }

  // ---------------- Phase 1: per-patch 4-qubit circuit ----------------
  for (int p = tid; p < ROWS * NPATCH; p += NTHREADS) {
    int r    = p / NPATCH;
    int pi   = p - r * NPATCH;
    int prow = pi / 14;
    int pcol = pi - prow * 14;
    int b    = b0 + r;

    float a0 = 0.f, a1 = 0.f, a2 = 0.f, a3 = 0.f;
    if (b < Btotal) {
      const float* img = x + (size_t)b * 784 + prow * 56 + pcol * 2;
      a0 = img[0]; a1 = img[1]; a2 = img[28]; a3 = img[29];
    }

    // encoding: product state  e_q = (cos(a/2), sin(a/2))
    float ec0 = __cosf(0.5f * a0), es0 = __sinf(0.5f * a0);
    float ec1 = __cosf(0.5f * a1), es1 = __sinf(0.5f * a1);
    float ec2 = __cosf(0.5f * a2), es2 = __sinf(0.5f * a2);
    float ec3 = __cosf(0.5f * a3), es3 = __sinf(0.5f * a3);

    float p01[4] = { ec0 * ec1, ec0 * es1, es0 * ec1, es0 * es1 };
    float p23[4] = { ec2 * ec3, ec2 * es3, es2 * ec3, es2 * es3 };

    float st[16];  // idx = q0*8 + q1*4 + q2*2 + q3
#pragma unroll
    for (int i = 0; i < 16; ++i) st[i] = p01[i >> 2] * p23[i & 3];

    // 3 layers of weight RYs + CNOT chain (all in registers)
#pragma unroll
    for (int l = 0; l < 3; ++l) {
      apply_ry<8>(st, cw[l * 4 + 0], sw[l * 4 + 0]);
      apply_ry<4>(st, cw[l * 4 + 1], sw[l * 4 + 1]);
      apply_ry<2>(st, cw[l * 4 + 2], sw[l * 4 + 2]);
      apply_ry<1>(st, cw[l * 4 + 3], sw[l * 4 + 3]);
      apply_cnot<8, 4>(st);
      apply_cnot<4, 2>(st);
      apply_cnot<2, 1>(st);
    }

    // <Z_q> expectation values
    float ev0 = 0.f, ev1 = 0.f, ev2 = 0.f, ev3 = 0.f;
#pragma unroll
    for (int i = 0; i < 16; ++i) {
      float pr = st[i] * st[i];
      ev0 += (i & 8) ? -pr : pr;
      ev1 += (i & 4) ? -pr : pr;
      ev2 += (i & 2) ? -pr : pr;
      ev3 += (i & 1) ? -pr : pr;
    }

    int fo = pi * 4;
    sFeats[r][fo + 0] = ev0;
    sFeats[r][fo + 1] = ev1;
    sFeats[r][fo + 2] = ev2;
    sFeats[r][fo + 3] = ev3;
  }

  __syncthreads();

  // ---------------- Phase 2: WMMA GEMM  [16 x 784] x [784 x 16] ----------------
  const int wave = tid >> 5;
  const int lane = tid & 31;
  const int mrow = lane & 15;                 // A row / B col for this lane
  const int ksel = (lane < 16) ? 0 : 2;       // K pair held by this lane half

  v8f acc = {};
  if (useBpad) {
    // v2f element offset for step kt: ((wave*49+kt)*2 + ksel/2)*16 + mrow
    const v2f* bptr = (const v2f*)bpad +
                      ((wave * KT_WAVE * 2) + (ksel >> 1)) * 16 + mrow;
    for (int kt = 0; kt < KT_WAVE; ++kt) {
      const int k0 = (wave * KT_WAVE + kt) * 4;
      v2f afrag = *(const v2f*)(&sFeats[mrow][k0 + ksel]);
      v2f bfrag = bptr[kt * 32];
      acc = __builtin_amdgcn_wmma_f32_16x16x4_f32(
          false, afrag, false, bfrag, (short)0, acc, false, false);
    }
  } else {
    // branchless fallback: clamp row, mask with multiply (no EXEC churn)
    const int   brow = (mrow < 10) ? mrow : 0;
    const float bm   = (mrow < 10) ? 1.0f : 0.0f;
    const float* wbase = lin_w + brow * KFEAT + ksel;
    for (int kt = 0; kt < KT_WAVE; ++kt) {
      const int k0 = (wave * KT_WAVE + kt) * 4;
      v2f afrag = *(const v2f*)(&sFeats[mrow][k0 + ksel]);
      v2f bfrag;
      bfrag.x = wbase[k0]     * bm;
      bfrag.y = wbase[k0 + 1] * bm;
      acc = __builtin_amdgcn_wmma_f32_16x16x4_f32(
          false, afrag, false, bfrag, (short)0, acc, false, false);
    }
  }

  // spill partial C tiles
#pragma unroll
  for (int i = 0; i < 8; ++i) {
    int row = i + ((lane < 16) ? 0 : 8);
    sC[wave][row][mrow] = acc[i];
  }
  __syncthreads();

  // reduce 4 partial tiles
  for (int e = tid; e < 256; e += NTHREADS) {
    int row = e >> 4, col = e & 15;
    float v = sC[0][row][col] + sC[1][row][col] + sC[2][row][col] + sC[3][row][col];
    sC[0][row][col] = v;
  }
  __syncthreads();

  // ---------------- Phase 3: bias + log_softmax, write out ----------------
  if (tid < ROWS) {
    int b = b0 + tid;
    if (b < Btotal) {
      float v[10];
      float m = -1e30f;
#pragma unroll
      for (int c = 0; c < 10; ++c) {
        v[c] = sC[0][tid][c] + lin_b[c];
        m = fmaxf(m, v[c]);
      }
      float sum = 0.f;
#pragma unroll
      for (int c = 0; c < 10; ++c) sum += __expf(v[c] - m);
      float lse = m + __logf(sum);
      float* op = out + (size_t)b * 10;
#pragma unroll
      for (int c = 0; c < 10; ++c) op[c] = v[c] - lse;
    }
  }
}

extern "C" void kernel_launch(void* const* d_in, const int* in_sizes, int n_in,
                              void* d_out, int out_size, void* d_ws, size_t ws_size,
                              hipStream_t stream) {
  const float* x     = (const float*)d_in[0];  // [B,1,28,28]
  const float* wts   = (const float*)d_in[1];  // [3,4]
  const float* lin_w = (const float*)d_in[2];  // [10,784]
  const float* lin_b = (const float*)d_in[3];  // [10]
  float* out = (float*)d_out;

  int Btotal = in_sizes[0] / 784;
  int grid   = (Btotal + ROWS - 1) / ROWS;

  int useBpad = (ws_size >= (size_t)BPAD_FLOATS * sizeof(float)) ? 1 : 0;
  if (useBpad) {
    prep_bpad_kernel<<<(BPAD_FLOATS + 255) / 256, 256, 0, stream>>>(
        lin_w, (float*)d_ws);
  }

  quanv_fused_kernel<<<grid, NTHREADS, 0, stream>>>(
      x, wts, lin_w, lin_b, (const float*)d_ws, useBpad, out, Btotal);
}